// GNN_19799799234858
// MI455X (gfx1250) — compile-verified
//
#include <hip/hip_runtime.h>

// LightGCN propagation on MI455X (gfx1250).
// Memory-bound SpMM: no dense GEMM exists in this model, so WMMA is not the
// right tool. CDNA5 paths used: TDM tensor_load_to_lds staging of packed edge
// tuples (TENSORcnt / s_wait_tensorcnt), ds_load_b128 tile reads, 128-bit
// gathers, and wave32 global_atomic_add_f32 scatter.

#define N_NODES   100000
#define EMB_DIM   64
#define N_EDGES   1250000
#define N_LAYERS  3
#define TILE      256            // edges per block in the scatter kernel

typedef __attribute__((ext_vector_type(4))) unsigned int v4u;
typedef __attribute__((ext_vector_type(8))) int          v8i;
typedef __attribute__((ext_vector_type(4))) int          v4i;

#if defined(__has_builtin)
#  if __has_builtin(__builtin_amdgcn_tensor_load_to_lds) && \
      __has_builtin(__builtin_amdgcn_s_wait_tensorcnt)
#    define USE_TDM 1
#  endif
#endif
#ifndef USE_TDM
#  define USE_TDM 0
#endif

#if USE_TDM
// Toolchain-dependent arity (probe-verified): ROCm 7.2 clang-22 -> 5 args;
// amdgpu-toolchain clang-23 -> 6 args (extra int32x8 before cpol).
__device__ __forceinline__ void tdm_load_to_lds(v4u g0, v8i g1) {
  v4i gz4 = {0, 0, 0, 0};
#if __clang_major__ >= 23
  v8i gz8 = {0, 0, 0, 0, 0, 0, 0, 0};
  __builtin_amdgcn_tensor_load_to_lds(g0, g1, gz4, gz4, gz8, 0);
#else
  __builtin_amdgcn_tensor_load_to_lds(g0, g1, gz4, gz4, 0);
#endif
}

// Opaque 128-bit LDS read: keeps the TDM-written tile from being folded to
// undef (no C++ stores alias it) and guarantees real ds_load_b128 traffic.
__device__ __forceinline__ v4i lds_read_b128(unsigned byte_off) {
  v4i r;
  asm volatile("ds_load_b128 %0, %1\n\t"
               "s_wait_dscnt 0x0"
               : "=v"(r) : "v"(byte_off) : "memory");
  return r;
}
#endif

// ----------------------------------------- scatter: y[col] += w * x[row] ----
__global__ __launch_bounds__(256) void k_scatter(const float* __restrict__ x,
                                                 float* __restrict__ y,
                                                 const int4* __restrict__ edata,
                                                 int nEdges) {
  __shared__ int4 s_tile[TILE];
  const int t        = threadIdx.x;
  const int tileBase = blockIdx.x * TILE;
  int nLocal = nEdges - tileBase;
  if (nLocal > TILE) nLocal = TILE;

  // LDS byte offset of the tile base (flat-address low 32 bits == LDS offset).
  const unsigned ldsBase = (unsigned)(unsigned long long)(const void*)&s_tile[0];

#if USE_TDM
  if (t < 32) {  // one wave issues the block's TDM DMA for the whole tile
    unsigned long long ga = (unsigned long long)(const void*)(edata + tileBase);
    unsigned td0 = (unsigned)nLocal * 4u;        // valid dwords; OOB reads -> 0
    v4u g0;
    g0.x = 1u;                                   // count=1, user descriptor
    g0.y = ldsBase;                              // lds_addr (bytes)
    g0.z = (unsigned)(ga & 0xFFFFFFFFu);         // global_addr[31:0]
    g0.w = 0x80000000u |                         // type=2 ("image")
           (unsigned)((ga >> 32) & 0x01FFFFFFu); // global_addr[56:32]
    v8i g1;
    g1[0] = 0x00020000;                          // data_size=4B; no pad/iter/mcast
    g1[1] = (int)((td0 & 0xFFFFu) << 16);        // tensor_dim0[15:0]
    g1[2] = (int)(((td0 >> 16) & 0xFFFFu) |      // tensor_dim0[31:16]
                  (1u << 16));                   // tensor_dim1 = 1
    g1[3] = (int)(1024u << 16);                  // tile_dim0 = 1024 dwords (4KB)
    g1[4] = 1;                                   // tile_dim1 = 1
    g1[5] = 1024;                                // tensor_dim0_stride
    g1[6] = 0;
    g1[7] = 0;
    tdm_load_to_lds(g0, g1);
    __builtin_amdgcn_s_wait_tensorcnt(0);
  }
  __syncthreads();
#else
  if (t < nLocal) s_tile[t] = edata[tileBase + t];
  __syncthreads();
#endif

  // 16 lanes per edge, float4 per lane -> 256B coalesced gather per edge.
  const int sub = t & 15;   // which 4-float chunk of the 64-dim row
  const int grp = t >> 4;   // edge group within the tile
  for (int k = grp; k < nLocal; k += 16) {
#if USE_TDM
    const v4i  ed = lds_read_b128(ldsBase + (unsigned)k * 16u);
    const int   r = ed.x;
    const int   c = ed.y;
    const float w = __int_as_float(ed.z);
#else
    const int4 ed = s_tile[k];
    const int   r = ed.x;
    const int   c = ed.y;
    const float w = __int_as_float(ed.z);
#endif
    const float4 v = *reinterpret_cast<const float4*>(
        x + (size_t)r * EMB_DIM + sub * 4);
    float* dst = y + (size_t)c * EMB_DIM + sub * 4;
    atomicAdd(dst + 0, w * v.x);
    atomicAdd(dst + 1, w * v.y);
    atomicAdd(dst + 2, w * v.z);
    atomicAdd(dst + 3, w * v.w);
  }
}

// ---------------------------------------------------------------- init ------
// i indexes float4 chunks of the 6.4M-float embedding table.
__global__ __launch_bounds__(256) void k_init(const float4* __restrict__ emb,
                                              float4* __restrict__ xcur,
                                              float4* __restrict__ acc,
                                              float* __restrict__ deg,
                                              int nVec) {
  int i = blockIdx.x * blockDim.x + threadIdx.x;
  if (i < nVec) {
    float4 v = emb[i];
    xcur[i] = v;
    acc[i]  = v;
  }
  if (i < N_NODES) deg[i] = 0.0f;
}

// ------------------------------------------------------------- degrees ------
__global__ __launch_bounds__(256) void k_deg(const long long* __restrict__ eidx,
                                             float* __restrict__ deg, int nE) {
  int e = blockIdx.x * blockDim.x + threadIdx.x;
  if (e < nE) {
    int c = (int)eidx[(size_t)nE + e];   // col = edge_index[1][e]
    atomicAdd(&deg[c], 1.0f);
  }
}

__global__ __launch_bounds__(256) void k_dinv(float* __restrict__ deg, int n) {
  int i = blockIdx.x * blockDim.x + threadIdx.x;
  if (i < n) {
    float d = deg[i];
    deg[i] = (d > 0.0f) ? rsqrtf(d) : 0.0f;
  }
}

// ------------------------------------------- pack {row, col, norm} 16B ------
__global__ __launch_bounds__(256) void k_pack(const long long* __restrict__ eidx,
                                              const float* __restrict__ dinv,
                                              int4* __restrict__ edata, int nE) {
  int e = blockIdx.x * blockDim.x + threadIdx.x;
  if (e < nE) {
    int r = (int)eidx[e];
    int c = (int)eidx[(size_t)nE + e];
    float w = dinv[r] * dinv[c];
    edata[e] = make_int4(r, c, __float_as_int(w), 0);
  }
}

// ---------------------------------------------------------------- zero ------
__global__ __launch_bounds__(256) void k_zero(float4* __restrict__ p, int nVec) {
  int i = blockIdx.x * blockDim.x + threadIdx.x;
  if (i < nVec) p[i] = make_float4(0.f, 0.f, 0.f, 0.f);
}

// ------------------------------------------------------- acc += x (*s) ------
__global__ __launch_bounds__(256) void k_acc(float4* __restrict__ acc,
                                             const float4* __restrict__ x,
                                             float scale, int nVec) {
  int i = blockIdx.x * blockDim.x + threadIdx.x;
  if (i < nVec) {
    float4 a = acc[i];
    float4 b = x[i];
    acc[i] = make_float4((a.x + b.x) * scale, (a.y + b.y) * scale,
                         (a.z + b.z) * scale, (a.w + b.w) * scale);
  }
}

// --------------------------------------------------------------- launch -----
extern "C" void kernel_launch(void* const* d_in, const int* in_sizes, int n_in,
                              void* d_out, int out_size, void* d_ws, size_t ws_size,
                              hipStream_t stream) {
  const float*     emb  = (const float*)d_in[0];
  const long long* eidx = (const long long*)d_in[1];   // int64 [2, E] row-major
  float* out = (float*)d_out;

  const size_t nv  = (size_t)N_NODES * EMB_DIM;        // 6.4M floats
  const int    nv4 = (int)(nv / 4);                    // 1.6M float4s
  char*  ws    = (char*)d_ws;
  float* xA    = (float*)ws;                           // 25.6 MB
  float* xB    = xA + nv;                              // 25.6 MB
  float* dinv  = xB + nv;                              // 0.4 MB (deg -> 1/sqrt(deg))
  int4*  edata = (int4*)(dinv + N_NODES);              // 20 MB, 16B-aligned

  const int gridV4 = (nv4 + 255) / 256;                // 6250
  const int gridE  = (N_EDGES + 255) / 256;            // 4883
  const int gridN  = (N_NODES + 255) / 256;            // 391

  k_init<<<gridV4, 256, 0, stream>>>((const float4*)emb, (float4*)xA,
                                     (float4*)out, dinv, nv4);
  k_deg <<<gridE,  256, 0, stream>>>(eidx, dinv, N_EDGES);
  k_dinv<<<gridN,  256, 0, stream>>>(dinv, N_NODES);
  k_pack<<<gridE,  256, 0, stream>>>(eidx, dinv, edata, N_EDGES);

  float* xc = xA;
  float* xn = xB;
  for (int l = 0; l < N_LAYERS; ++l) {
    k_zero   <<<gridV4, 256, 0, stream>>>((float4*)xn, nv4);
    k_scatter<<<gridE,  256, 0, stream>>>(xc, xn, edata, N_EDGES);
    const float scale = (l == N_LAYERS - 1) ? (1.0f / (N_LAYERS + 1)) : 1.0f;
    k_acc    <<<gridV4, 256, 0, stream>>>((float4*)out, (const float4*)xn,
                                          scale, nv4);
    float* tmp = xc; xc = xn; xn = tmp;
  }
}